// EmbedderCRFModule_55731495633397
// MI455X (gfx1250) — compile-verified
//
#include <hip/hip_runtime.h>
#include <hip/hip_bf16.h>
#include <math.h>

typedef float v2f __attribute__((ext_vector_type(2)));
typedef float v8f __attribute__((ext_vector_type(8)));

#define Bb   64
#define Tsub 512
#define Ll   256
#define Hh   1024
#define Kk   9

#define LOGITS_OFF 1
#define PRED_OFF   (1 + Bb * Ll * Kk)

// ---------------------------------------------------------------------------
// Kernel 1: gather rows of sequence_output via offsets, multiply by the
// (H x 9) classifier (padded to 16 cols), accumulate with V_WMMA_F32_16X16X4_F32.
// A-tiles are streamed into double-buffered LDS with the CDNA5 async
// global->LDS path (GLOBAL_LOAD_ASYNC_TO_LDS_B128, ASYNCcnt), overlapping the
// next chunk's HBM traffic with the current chunk's WMMAs.
// One wave (32 threads) per 16-row output tile. 1024 tiles.
// ---------------------------------------------------------------------------
__global__ __launch_bounds__(32)
void gemm_logits_kernel(const float* __restrict__ seq,
                        const int*   __restrict__ offsets,
                        const float* __restrict__ cw,   // (H, 9) row-major
                        const float* __restrict__ cb,   // (9)
                        float*       __restrict__ out)  // d_out base
{
    // stride 68 floats: 16B-aligned rows for B128 async writes; bank =
    // (4*row + k) % 64 -> conflict-free across the 16 rows per half-wave.
    __shared__ __align__(16) float As[2][16][68];
    __shared__ __align__(16) float Ws[2][64][16];
    __shared__ unsigned rowStart[16];   // element offset of each gathered row

    const int lane    = threadIdx.x;
    const int tileRow = blockIdx.x * 16;

    if (lane < 16) {
        int r   = tileRow + lane;
        int b   = r >> 8;          // / L (=256)
        int l   = r & 255;         // % L
        int off = offsets[b * Ll + l];
        rowStart[lane] = (unsigned)((b * Tsub + off) * Hh);
    }
    __syncthreads();

    const int c4    = lane & 15;   // float4 column within a 64-wide chunk
    const int rbase = lane >> 4;   // 0 or 1

    // Issue 8 async B128 global->LDS copies (one 16x64 A chunk) per lane.
    auto issueA = [&](int bufi, int h0) {
        #pragma unroll
        for (int i = 0; i < 8; ++i) {
            int row = rbase + 2 * i;
            const float* g   = seq + rowStart[row] + h0 + c4 * 4;
            unsigned     lds = (unsigned)(unsigned long long)&As[bufi][row][c4 * 4];
            asm volatile("global_load_async_to_lds_b128 %0, %1, off"
                         :: "v"(lds), "v"((unsigned long long)g)
                         : "memory");
        }
    };
    // Synchronous staging of a 64x16 zero-padded W chunk (L2-resident).
    auto loadW = [&](int bufi, int h0) {
        #pragma unroll
        for (int i = 0; i < 8; ++i) {
            int kk = rbase + 2 * i;
            #pragma unroll
            for (int j = 0; j < 4; ++j) {
                int kr = kk + 16 * j;            // 0..63
                float w = 0.0f;
                if (c4 < Kk) w = cw[(h0 + kr) * Kk + c4];
                Ws[bufi][kr][c4] = w;
            }
        }
    };

    issueA(0, 0);
    loadW(0, 0);

    v8f acc = {};
    const int m    = lane & 15;          // A row / B,C column
    const int koff = (lane >> 4) * 2;    // K sub-offset per half-wave

    for (int c = 0; c < 16; ++c) {
        const int nxt = c + 1;
        if (nxt < 16) {
            issueA(nxt & 1, nxt * 64);   // prefetch next chunk into other buffer
            loadW(nxt & 1, nxt * 64);
            asm volatile("s_wait_asynccnt 0x8" ::: "memory");  // prev chunk landed
        } else {
            asm volatile("s_wait_asynccnt 0x0" ::: "memory");  // drain
        }
        __syncthreads();

        const float (*A)[68] = As[c & 1];
        const float (*W)[16] = Ws[c & 1];
        #pragma unroll
        for (int kk = 0; kk < 64; kk += 4) {
            v2f a, b;
            a.x = A[m][kk + koff + 0];
            a.y = A[m][kk + koff + 1];
            b.x = W[kk + koff + 0][m];
            b.y = W[kk + koff + 1][m];
            acc = __builtin_amdgcn_wmma_f32_16x16x4_f32(
                false, a, false, b, (short)0, acc, false, false);
        }
        __syncthreads();
    }

    // ---- epilogue: C layout = VGPR v -> M = v (+8 for upper lanes), N = lane&15
    const int col   = lane & 15;
    const int mbase = (lane < 16) ? 0 : 8;
    if (col < Kk) {
        float bias = cb[col];
        #pragma unroll
        for (int v = 0; v < 8; ++v) {
            int r = tileRow + mbase + v;
            out[LOGITS_OFF + r * Kk + col] = acc[v] + bias;
        }
    }
}

// ---------------------------------------------------------------------------
// Kernel 2a: zero the loss accumulator
// ---------------------------------------------------------------------------
__global__ void init_loss_kernel(float* out) { out[0] = 0.0f; }

// ---------------------------------------------------------------------------
// Kernel 2b: CRF log-likelihood. One wave per batch element; lanes 0..8 hold
// the 9 states, cross-lane shuffles implement the 9-wide logsumexp.
// ---------------------------------------------------------------------------
__global__ __launch_bounds__(32)
void crf_forward_kernel(const float*         __restrict__ logits,
                        const unsigned char* __restrict__ mask,
                        const int*           __restrict__ labels,
                        const float*         __restrict__ start,
                        const float*         __restrict__ endt,
                        const float*         __restrict__ trans,
                        float*               __restrict__ loss_out)
{
    const int b    = blockIdx.x;
    const int lane = threadIdx.x;
    const float*         em = logits + (size_t)b * Ll * Kk;
    const unsigned char* mb = mask   + b * Ll;
    const int*           lb = labels + b * Ll;

    // per-lane transition column trans[i][lane]
    float tc[Kk];
    #pragma unroll
    for (int i = 0; i < Kk; ++i)
        tc[i] = (lane < Kk) ? trans[i * Kk + lane] : 0.0f;

    float alpha = -1e30f;
    if (lane < Kk) alpha = start[lane] + em[lane];

    for (int t = 1; t < Ll; ++t) {
        float emj = (lane < Kk) ? em[t * Kk + lane] : 0.0f;
        int   m   = mb[t];
        float av[Kk];
        #pragma unroll
        for (int i = 0; i < Kk; ++i) av[i] = __shfl(alpha, i, 32);
        if (lane < Kk) {
            float mx = -1e30f;
            #pragma unroll
            for (int i = 0; i < Kk; ++i) {
                float v = av[i] + tc[i];
                av[i] = v;
                if (v > mx) mx = v;
            }
            float s = 0.0f;
            #pragma unroll
            for (int i = 0; i < Kk; ++i) s += __expf(av[i] - mx);
            float nxt = mx + __logf(s) + emj;
            alpha = m ? nxt : alpha;
        }
    }

    float fin[Kk];
    #pragma unroll
    for (int i = 0; i < Kk; ++i) fin[i] = __shfl(alpha, i, 32);

    if (lane == 0) {
        // denominator = logsumexp(alpha_L + end)
        float mx = -1e30f;
        #pragma unroll
        for (int i = 0; i < Kk; ++i) {
            fin[i] += endt[i];
            if (fin[i] > mx) mx = fin[i];
        }
        float s = 0.0f;
        #pragma unroll
        for (int i = 0; i < Kk; ++i) s += __expf(fin[i] - mx);
        float denom = mx + __logf(s);

        // gold-path score (serial; K=9 so all lookups are L2-resident)
        // tags = labels * mask  (elementwise over the whole row)
        int   prev  = lb[0] * (int)(mb[0] != 0);
        float score = start[prev] + em[prev];
        int   nmask = (mb[0] != 0);
        for (int t = 1; t < Ll; ++t) {
            int tag = lb[t] * (int)(mb[t] != 0);
            float contrib = trans[prev * Kk + tag] + em[t * Kk + tag];
            if (mb[t]) score += contrib;
            nmask += (mb[t] != 0);
            prev = tag;
        }
        int seq_end  = nmask - 1;
        int last_tag = lb[seq_end] * (int)(mb[seq_end] != 0);
        score += endt[last_tag];

        float llh = score - denom;
        atomicAdd(loss_out, -llh * (1.0f / (float)Bb));
    }
}

// ---------------------------------------------------------------------------
// Kernel 3: Viterbi decode. One wave per batch element; backpointer history
// lives in LDS (255 x 9 ints); lane 0 backtracks and writes predicts.
// ---------------------------------------------------------------------------
__global__ __launch_bounds__(32)
void crf_viterbi_kernel(const float*         __restrict__ logits,
                        const unsigned char* __restrict__ mask,
                        const float*         __restrict__ start,
                        const float*         __restrict__ endt,
                        const float*         __restrict__ trans,
                        float*               __restrict__ pred_out)
{
    __shared__ int hist[(Ll - 1) * Kk];

    const int b    = blockIdx.x;
    const int lane = threadIdx.x;
    const float*         em = logits + (size_t)b * Ll * Kk;
    const unsigned char* mb = mask   + b * Ll;

    float tc[Kk];
    #pragma unroll
    for (int i = 0; i < Kk; ++i)
        tc[i] = (lane < Kk) ? trans[i * Kk + lane] : 0.0f;

    float score = (lane < Kk) ? (start[lane] + em[lane]) : -1e30f;

    for (int t = 1; t < Ll; ++t) {
        float sv[Kk];
        #pragma unroll
        for (int i = 0; i < Kk; ++i) sv[i] = __shfl(score, i, 32);
        if (lane < Kk) {
            float best = -1e30f;
            int   bi   = 0;
            #pragma unroll
            for (int i = 0; i < Kk; ++i) {
                float v = sv[i] + tc[i];
                if (v > best) { best = v; bi = i; }   // strict > == jnp.argmax tie rule
            }
            int   m   = mb[t];
            float nxt = best + em[t * Kk + lane];
            score = m ? nxt : score;
            hist[(t - 1) * Kk + lane] = m ? bi : lane;
        }
    }
    __syncthreads();

    float sv[Kk];
    #pragma unroll
    for (int i = 0; i < Kk; ++i) sv[i] = __shfl(score, i, 32);

    if (lane == 0) {
        float best = -1e30f;
        int   tag  = 0;
        #pragma unroll
        for (int i = 0; i < Kk; ++i) {
            float v = sv[i] + endt[i];
            if (v > best) { best = v; tag = i; }
        }
        pred_out[b * Ll + (Ll - 1)] = mb[Ll - 1] ? (float)tag : 0.0f;
        for (int t = Ll - 2; t >= 0; --t) {
            tag = hist[t * Kk + tag];
            pred_out[b * Ll + t] = mb[t] ? (float)tag : 0.0f;
        }
    }
}

// ---------------------------------------------------------------------------
extern "C" void kernel_launch(void* const* d_in, const int* in_sizes, int n_in,
                              void* d_out, int out_size, void* d_ws, size_t ws_size,
                              hipStream_t stream)
{
    const float*         seq     = (const float*)d_in[0];
    // d_in[1] = attention_mask (unused by the reference computation)
    const int*           offsets = (const int*)d_in[2];
    const unsigned char* mask    = (const unsigned char*)d_in[3];
    const int*           labels  = (const int*)d_in[4];
    const float*         cw      = (const float*)d_in[5];
    const float*         cb      = (const float*)d_in[6];
    const float*         st      = (const float*)d_in[7];
    const float*         en      = (const float*)d_in[8];
    const float*         tr      = (const float*)d_in[9];
    float*               out     = (float*)d_out;

    gemm_logits_kernel<<<(Bb * Ll) / 16, 32, 0, stream>>>(seq, offsets, cw, cb, out);
    init_loss_kernel<<<1, 1, 0, stream>>>(out);
    crf_forward_kernel<<<Bb, 32, 0, stream>>>(out + LOGITS_OFF, mask, labels, st, en, tr, out);
    crf_viterbi_kernel<<<Bb, 32, 0, stream>>>(out + LOGITS_OFF, mask, st, en, tr, out + PRED_OFF);
}